// RandomWalkPE_38654705664483
// MI455X (gfx1250) — compile-verified
//
#include <hip/hip_runtime.h>
#include <math.h>

// ---------------------------------------------------------------------------
// RandomWalkPE + 2-layer GELU MLP for MI455X (gfx1250, wave32, WMMA)
//
// Phase 1 (L2-bound):  deg count -> deg_inv -> 16x {prep, scatter-add}
// Phase 2 (WMMA):      pe[100000x16] @ W1[16x64] -> GELU -> @ W2[64x64]
//                      using V_WMMA_F32_16X16X4_F32 (f32 in / f32 acc)
// ---------------------------------------------------------------------------

typedef __attribute__((ext_vector_type(2))) float v2f;
typedef __attribute__((ext_vector_type(8))) float v8f;

#define NN 100000   // num_nodes (matches reference constant)
#define HID 64
#define WLEN 16
#define LDS_STRIDE 66   // 64 + 2 pad: avoids bank conflicts on transposed reads

// ------------------------------- utility kernels ---------------------------

__global__ void rwpe_zero_kernel(unsigned* __restrict__ p, long long count) {
    long long i = (long long)blockIdx.x * blockDim.x + threadIdx.x;
    if (i < count) p[i] = 0u;
}

__global__ void rwpe_deg_kernel(const int* __restrict__ col,
                                unsigned* __restrict__ deg, int E) {
    int e = blockIdx.x * blockDim.x + threadIdx.x;
    if (e < E) atomicAdd(&deg[col[e]], 1u);
}

// in-place: reinterpret deg counts as floats deg_inv = 1/max(deg,1)
__global__ void rwpe_deginv_kernel(unsigned* deg_u, float* deg_inv, int N) {
    int i = blockIdx.x * blockDim.x + threadIdx.x;
    if (i < N) {
        float d = (float)deg_u[i];
        deg_inv[i] = 1.0f / fmaxf(d, 1.0f);
    }
}

// pd[n] = prev[n] * deg_inv[n]   (prev == nullptr means walk_dist == ones)
__global__ void rwpe_prep_kernel(const float* __restrict__ prev,
                                 const float* __restrict__ deg_inv,
                                 float* __restrict__ pd, int N) {
    int i = blockIdx.x * blockDim.x + threadIdx.x;
    if (i < N) {
        float p = prev ? prev[i] : 1.0f;
        pd[i] = p * deg_inv[i];
    }
}

// dst[col[e]] += pd[row[e]]   (native global_atomic_add_f32)
__global__ void rwpe_scatter_kernel(const int* __restrict__ row,
                                    const int* __restrict__ col,
                                    const float* __restrict__ pd,
                                    float* __restrict__ dst, int E) {
    int e = blockIdx.x * blockDim.x + threadIdx.x;
    if (e < E) {
        float v = pd[row[e]];
        unsafeAtomicAdd(&dst[col[e]], v);
    }
}

// ------------------------------- WMMA MLP ----------------------------------
// One wave handles 16 nodes. pe is stored step-major: pe[k*NN + node],
// which is exactly A[M=node][K=step] in column-of-memory order.
//
// A 16x4 f32 layout : a[j] = A[M=n][K = kbase + 2*half + j]
// B 4x16  layout    : b[j] = B[K = kbase + 2*half + j][N = n]
// C/D 16x16 layout  : acc[v] = C[M = v + 8*half][N = n]
//   with lane = half*16 + n, half in {0,1}, n in {0..15}

__device__ __forceinline__ v8f wmma_f32_4(v2f a, v2f b, v8f c) {
    return __builtin_amdgcn_wmma_f32_16x16x4_f32(
        /*neg_a=*/false, a, /*neg_b=*/false, b,
        /*c_mod=*/(short)0, c, /*reuse_a=*/false, /*reuse_b=*/false);
}

__global__ __launch_bounds__(256) void rwpe_mlp_kernel(
    const float* __restrict__ pe,   // [WLEN][NN] step-major
    const float* __restrict__ W1,   // [16][64] row-major
    const float* __restrict__ b1,   // [64]
    const float* __restrict__ W2,   // [64][64] row-major
    const float* __restrict__ b2,   // [64]
    float* __restrict__ out,        // [NN][64] row-major
    int nTiles) {
    __shared__ float lds[8][16 * LDS_STRIDE];

    const int lane = threadIdx.x & 31;
    const int wave = threadIdx.x >> 5;
    const int tile = blockIdx.x * 8 + wave;
    if (tile >= nTiles) return;           // wave-uniform: EXEC stays all-1s

    const int nb   = tile * 16;           // base node of this 16-row tile
    const int half = lane >> 4;
    const int n    = lane & 15;
    float* hbuf = lds[wave];

    // ---- GEMM1: h = GELU(pe @ W1 + b1), staged into LDS ----
    #pragma unroll
    for (int nt = 0; nt < 4; ++nt) {
        const float bias = b1[nt * 16 + n];
        v8f acc = {bias, bias, bias, bias, bias, bias, bias, bias};
        #pragma unroll
        for (int kb = 0; kb < 4; ++kb) {
            const int k0 = 4 * kb + 2 * half;
            v2f a, b;
            a.x = pe[(long long)(k0 + 0) * NN + nb + n];
            a.y = pe[(long long)(k0 + 1) * NN + nb + n];
            b.x = W1[(k0 + 0) * HID + nt * 16 + n];
            b.y = W1[(k0 + 1) * HID + nt * 16 + n];
            acc = wmma_f32_4(a, b, acc);
        }
        #pragma unroll
        for (int v = 0; v < 8; ++v) {
            float x = acc[v];
            // exact GELU: 0.5*x*(1+erf(x/sqrt(2)))
            float g = 0.5f * x * (1.0f + erff(x * 0.70710678118654752f));
            hbuf[(v + 8 * half) * LDS_STRIDE + nt * 16 + n] = g;
        }
    }
    // Intra-wave LDS RAW: DS ops from one wave execute in order; compiler
    // inserts the required s_wait_dscnt for its own dependent loads.

    // ---- GEMM2: out = h @ W2 + b2 ----
    #pragma unroll
    for (int nt = 0; nt < 4; ++nt) {
        const float bias = b2[nt * 16 + n];
        v8f acc = {bias, bias, bias, bias, bias, bias, bias, bias};
        #pragma unroll
        for (int kb = 0; kb < 16; ++kb) {
            const int k0 = 4 * kb + 2 * half;
            v2f a, b;
            a.x = hbuf[n * LDS_STRIDE + k0 + 0];
            a.y = hbuf[n * LDS_STRIDE + k0 + 1];
            b.x = W2[(k0 + 0) * HID + nt * 16 + n];
            b.y = W2[(k0 + 1) * HID + nt * 16 + n];
            acc = wmma_f32_4(a, b, acc);
        }
        #pragma unroll
        for (int v = 0; v < 8; ++v) {
            out[(long long)(nb + v + 8 * half) * HID + nt * 16 + n] = acc[v];
        }
    }
}

// ------------------------------- launcher ----------------------------------

extern "C" void kernel_launch(void* const* d_in, const int* in_sizes, int n_in,
                              void* d_out, int out_size, void* d_ws, size_t ws_size,
                              hipStream_t stream) {
    const int*   edge = (const int*)d_in[0];       // [2][E]
    const int    E    = in_sizes[0] / 2;
    const int*   row  = edge;
    const int*   col  = edge + E;
    const float* W1   = (const float*)d_in[2];
    const float* b1   = (const float*)d_in[3];
    const float* W2   = (const float*)d_in[4];
    const float* b2   = (const float*)d_in[5];
    float*       out  = (float*)d_out;
    const int    N    = NN;

    // workspace layout (floats/uints, all 4B):
    //   [0 .. N)        deg counts (u32), then in-place deg_inv (f32)
    //   [N .. 2N)       pd  (prev * deg_inv per node)
    //   [2N .. 18N)     pe  [16][N] step-major
    float*    ws      = (float*)d_ws;
    unsigned* deg_u   = (unsigned*)ws;
    float*    deg_inv = ws;
    float*    pd      = ws + N;
    float*    pe      = ws + 2 * (long long)N;

    const int TB = 256;
    const long long zcount = 18LL * N;
    rwpe_zero_kernel<<<(int)((zcount + TB - 1) / TB), TB, 0, stream>>>(
        (unsigned*)d_ws, zcount);

    rwpe_deg_kernel<<<(E + TB - 1) / TB, TB, 0, stream>>>(col, deg_u, E);
    rwpe_deginv_kernel<<<(N + TB - 1) / TB, TB, 0, stream>>>(deg_u, deg_inv, N);

    for (int t = 0; t < WLEN; ++t) {
        const float* prev = (t == 0) ? nullptr : (pe + (long long)(t - 1) * N);
        rwpe_prep_kernel<<<(N + TB - 1) / TB, TB, 0, stream>>>(prev, deg_inv, pd, N);
        rwpe_scatter_kernel<<<(E + TB - 1) / TB, TB, 0, stream>>>(
            row, col, pd, pe + (long long)t * N, E);
    }

    const int nTiles = N / 16;  // 6250 (N divisible by 16)
    rwpe_mlp_kernel<<<(nTiles + 7) / 8, TB, 0, stream>>>(
        pe, W1, b1, W2, b2, out, nTiles);
}